// _SelectiveSSMBlock_49598282334579
// MI455X (gfx1250) — compile-verified
//
#include <hip/hip_runtime.h>
#include <hip/hip_bf16.h>
#include <math.h>

// Problem constants (match reference)
#define BSZ   2
#define LSEQ  2048
#define DDIM  1024
#define NST   16
#define MROWS (BSZ * LSEQ)     // 4096 rows total
#define LN_EPS 1e-5f

// Chunked-scan config: 32 chunks of 64 steps
#define CCH 32
#define TCH (LSEQ / CCH)

// TDM GEMM config
#define KCH 32                 // K-chunk staged per TDM transfer
#define LDS_STRIDE 36          // 32 data floats + 4 pad DWORDs (bank-conflict-free)

typedef __attribute__((ext_vector_type(2))) float v2f;
typedef __attribute__((ext_vector_type(8))) float v8f;
typedef __attribute__((ext_vector_type(4))) unsigned int u32x4;
typedef __attribute__((ext_vector_type(4))) int i32x4;
typedef __attribute__((ext_vector_type(8))) int i32x8;

#if __has_builtin(__builtin_amdgcn_tensor_load_to_lds)
#define HAVE_TDM 1
#else
#define HAVE_TDM 0
#endif

__device__ __forceinline__ float softplus_f(float v) {
    // jax.nn.softplus(x) == max(x,0) + log1p(exp(-|x|))
    return fmaxf(v, 0.f) + log1pf(__expf(-fabsf(v)));
}

// ---------------------------------------------------------------------------
// Kernel 1: LayerNorm. One block (256 threads) per row of D=1024.
// ---------------------------------------------------------------------------
__global__ __launch_bounds__(256) void ln_kernel(
    const float* __restrict__ x, const float* __restrict__ gamma,
    const float* __restrict__ beta, float* __restrict__ xn) {
    const int row = blockIdx.x;
    const float4 v = ((const float4*)(x + (size_t)row * DDIM))[threadIdx.x];

    float s  = v.x + v.y + v.z + v.w;
    float ss = v.x * v.x + v.y * v.y + v.z * v.z + v.w * v.w;

    __shared__ float r0[256];
    __shared__ float r1[256];
    r0[threadIdx.x] = s;
    r1[threadIdx.x] = ss;
    __syncthreads();
    for (int off = 128; off > 0; off >>= 1) {
        if (threadIdx.x < off) {
            r0[threadIdx.x] += r0[threadIdx.x + off];
            r1[threadIdx.x] += r1[threadIdx.x + off];
        }
        __syncthreads();
    }
    const float mu   = r0[0] * (1.f / DDIM);
    const float var  = r1[0] * (1.f / DDIM) - mu * mu;
    const float rstd = rsqrtf(var + LN_EPS);

    const float4 g = ((const float4*)gamma)[threadIdx.x];
    const float4 b = ((const float4*)beta)[threadIdx.x];
    float4 o;
    o.x = (v.x - mu) * rstd * g.x + b.x;
    o.y = (v.y - mu) * rstd * g.y + b.y;
    o.z = (v.z - mu) * rstd * g.z + b.z;
    o.w = (v.w - mu) * rstd * g.w + b.w;
    ((float4*)(xn + (size_t)row * DDIM))[threadIdx.x] = o;
}

// ---------------------------------------------------------------------------
// TDM helper: DMA a 2D tile (tile_rows x KCH f32, row stride DDIM elements)
// from global into LDS with +4-DWORD padding after every 32 DWORDs
// (-> LDS row stride = 36 floats). D# layout per CDNA5 ISA 8.3/8.4.
// ---------------------------------------------------------------------------
#if HAVE_TDM
__device__ __forceinline__ void tdm_load_2d(unsigned int lds_byte_addr,
                                            const void* gptr,
                                            unsigned int tile_rows,
                                            unsigned int tensor_rows) {
    const unsigned long long ga = (unsigned long long)(size_t)gptr;
    u32x4 g0;
    g0[0] = 1u;                                    // count=1, user descriptor
    g0[1] = lds_byte_addr;                         // lds_addr
    g0[2] = (unsigned int)(ga & 0xffffffffull);    // global_addr[31:0]
    g0[3] = (unsigned int)(ga >> 32) | (2u << 30); // global_addr[56:32] | type=2
    i32x8 g1;
    // data_size=4B (2), pad_enable, pad_interval=32 DWORDs (4), pad_amount=4 DWORDs (3)
    g1[0] = (int)((2u << 16) | (1u << 20) | (4u << 22) | (3u << 25));
    g1[1] = (int)((unsigned)DDIM << 16);                        // tensor_dim0 lo16
    g1[2] = (int)(tensor_rows << 16);                           // dim0 hi | tensor_dim1 lo16
    g1[3] = (int)((tensor_rows >> 16) | ((unsigned)KCH << 16)); // dim1 hi | tile_dim0
    g1[4] = (int)tile_rows;                                     // tile_dim1 | tile_dim2=0
    g1[5] = (int)DDIM;                                          // tensor_dim0_stride lo32
    g1[6] = 0;
    g1[7] = 0;
    const i32x4 z4 = {0, 0, 0, 0};
#if defined(__clang_major__) && __clang_major__ >= 23
    const i32x8 z8 = {0, 0, 0, 0, 0, 0, 0, 0};
    __builtin_amdgcn_tensor_load_to_lds(g0, g1, z4, z4, z8, 0);
#else
    __builtin_amdgcn_tensor_load_to_lds(g0, g1, z4, z4, 0);
#endif
}

// ---------------------------------------------------------------------------
// Kernel 2 (TDM): dt = softplus(xn @ W_dt^T + b_dt).
// Block = 8 waves = 128x64 output tile. Per 32-wide K-chunk, TDM stages the
// 128x32 xn tile and 64x32 W tile into double-buffered LDS; inner loop is
// ds_load + v_wmma_f32_16x16x4_f32 only. Grid: (DDIM/64, MROWS/128).
// ---------------------------------------------------------------------------
__global__ __launch_bounds__(256) void gemm_dt_tdm_kernel(
    const float* __restrict__ xn, const float* __restrict__ W,
    const float* __restrict__ bias, float* __restrict__ dt) {
    __shared__ float lds[13824];   // A: 2 x 128*36 | W: 2 x 64*36  (55.3 KB)
    const int lane  = threadIdx.x & 31;
    const int wave  = threadIdx.x >> 5;
    const int m0    = blockIdx.y * 128;
    const int n0    = blockIdx.x * 64;
    const int l15   = lane & 15;
    const int khalf = (lane >> 4) * 2;

    const unsigned ldsbase = (unsigned)(size_t)&lds[0];
    const unsigned aoff[2] = {0u, 4608u * 4u};
    const unsigned woff[2] = {9216u * 4u, 11520u * 4u};

    if (wave == 0) {
        tdm_load_2d(ldsbase + aoff[0], xn + (size_t)m0 * DDIM, 128u, (unsigned)MROWS);
        tdm_load_2d(ldsbase + woff[0], W + (size_t)n0 * DDIM, 64u, (unsigned)DDIM);
        __builtin_amdgcn_s_wait_tensorcnt(0);
    }
    __syncthreads();

    v8f acc0 = {}, acc1 = {}, acc2 = {}, acc3 = {};
    int buf = 0;
    for (int kc = 0; kc < DDIM; kc += KCH) {
        if (kc + KCH < DDIM && wave == 0) {
            tdm_load_2d(ldsbase + aoff[buf ^ 1],
                        xn + (size_t)m0 * DDIM + (kc + KCH), 128u, (unsigned)MROWS);
            tdm_load_2d(ldsbase + woff[buf ^ 1],
                        W + (size_t)n0 * DDIM + (kc + KCH), 64u, (unsigned)DDIM);
        }
        const float* As = &lds[aoff[buf] / 4] + (wave * 16 + l15) * LDS_STRIDE + khalf;
        const float* Ws = &lds[woff[buf] / 4] + l15 * LDS_STRIDE + khalf;
#pragma unroll
        for (int kk = 0; kk < KCH; kk += 4) {
            const v2f a  = *(const v2f*)(As + kk);
            const v2f b0 = *(const v2f*)(Ws + kk);
            const v2f b1 = *(const v2f*)(Ws + 16 * LDS_STRIDE + kk);
            const v2f b2 = *(const v2f*)(Ws + 32 * LDS_STRIDE + kk);
            const v2f b3 = *(const v2f*)(Ws + 48 * LDS_STRIDE + kk);
            acc0 = __builtin_amdgcn_wmma_f32_16x16x4_f32(false, a, false, b0, (short)0, acc0, false, false);
            acc1 = __builtin_amdgcn_wmma_f32_16x16x4_f32(false, a, false, b1, (short)0, acc1, false, false);
            acc2 = __builtin_amdgcn_wmma_f32_16x16x4_f32(false, a, false, b2, (short)0, acc2, false, false);
            acc3 = __builtin_amdgcn_wmma_f32_16x16x4_f32(false, a, false, b3, (short)0, acc3, false, false);
        }
        if (kc + KCH < DDIM) {
            if (wave == 0) __builtin_amdgcn_s_wait_tensorcnt(0);
            __syncthreads();
            buf ^= 1;
        }
    }

    const int rbase = m0 + wave * 16 + ((lane >> 4) * 8);
    const float bv0 = bias[n0 + l15];
    const float bv1 = bias[n0 + 16 + l15];
    const float bv2 = bias[n0 + 32 + l15];
    const float bv3 = bias[n0 + 48 + l15];
#pragma unroll
    for (int r = 0; r < 8; ++r) {
        float* out = dt + (size_t)(rbase + r) * DDIM + n0 + l15;
        out[0]  = softplus_f(acc0[r] + bv0);
        out[16] = softplus_f(acc1[r] + bv1);
        out[32] = softplus_f(acc2[r] + bv2);
        out[48] = softplus_f(acc3[r] + bv3);
    }
}
#endif  // HAVE_TDM

// ---------------------------------------------------------------------------
// Kernel 2 (fallback): direct-load WMMA GEMM.
// ---------------------------------------------------------------------------
__global__ __launch_bounds__(256) void gemm_dt_direct_kernel(
    const float* __restrict__ xn, const float* __restrict__ W,
    const float* __restrict__ bias, float* __restrict__ dt) {
    const int lane  = threadIdx.x & 31;
    const int wave  = threadIdx.x >> 5;
    const int m0    = blockIdx.y * 128 + wave * 16;
    const int n0    = blockIdx.x * 64;
    const int l15   = lane & 15;
    const int khalf = (lane >> 4) * 2;

    const float* arow = xn + (size_t)(m0 + l15) * DDIM + khalf;
    const float* wrow = W + (size_t)(n0 + l15) * DDIM + khalf;

    v8f acc0 = {}, acc1 = {}, acc2 = {}, acc3 = {};
#pragma unroll 4
    for (int k = 0; k < DDIM; k += 4) {
        const v2f a  = *(const v2f*)(arow + k);
        const v2f b0 = *(const v2f*)(wrow + k);
        const v2f b1 = *(const v2f*)(wrow + (size_t)16 * DDIM + k);
        const v2f b2 = *(const v2f*)(wrow + (size_t)32 * DDIM + k);
        const v2f b3 = *(const v2f*)(wrow + (size_t)48 * DDIM + k);
        acc0 = __builtin_amdgcn_wmma_f32_16x16x4_f32(false, a, false, b0, (short)0, acc0, false, false);
        acc1 = __builtin_amdgcn_wmma_f32_16x16x4_f32(false, a, false, b1, (short)0, acc1, false, false);
        acc2 = __builtin_amdgcn_wmma_f32_16x16x4_f32(false, a, false, b2, (short)0, acc2, false, false);
        acc3 = __builtin_amdgcn_wmma_f32_16x16x4_f32(false, a, false, b3, (short)0, acc3, false, false);
    }

    const int rbase = m0 + ((lane >> 4) * 8);
    const float bv0 = bias[n0 + l15];
    const float bv1 = bias[n0 + 16 + l15];
    const float bv2 = bias[n0 + 32 + l15];
    const float bv3 = bias[n0 + 48 + l15];
#pragma unroll
    for (int r = 0; r < 8; ++r) {
        float* out = dt + (size_t)(rbase + r) * DDIM + n0 + l15;
        out[0]  = softplus_f(acc0[r] + bv0);
        out[16] = softplus_f(acc1[r] + bv1);
        out[32] = softplus_f(acc2[r] + bv2);
        out[48] = softplus_f(acc3[r] + bv3);
    }
}

// ---------------------------------------------------------------------------
// Kernel 3: B_in = xn @ W_B^T, C_in = xn @ W_C^T  (N = 16 = one WMMA tile).
// ---------------------------------------------------------------------------
__global__ __launch_bounds__(256) void gemm_bc_kernel(
    const float* __restrict__ xn, const float* __restrict__ WB,
    const float* __restrict__ WC, float* __restrict__ Bin,
    float* __restrict__ Cin) {
    const int lane  = threadIdx.x & 31;
    const int wave  = threadIdx.x >> 5;
    const int m0    = (blockIdx.x * 8 + wave) * 16;
    const int l15   = lane & 15;
    const int khalf = (lane >> 4) * 2;

    const float* arow  = xn + (size_t)(m0 + l15) * DDIM + khalf;
    const float* wbrow = WB + (size_t)l15 * DDIM + khalf;
    const float* wcrow = WC + (size_t)l15 * DDIM + khalf;

    v8f accB = {}, accC = {};
#pragma unroll 4
    for (int k = 0; k < DDIM; k += 4) {
        const v2f a  = *(const v2f*)(arow + k);
        const v2f bb = *(const v2f*)(wbrow + k);
        const v2f bc = *(const v2f*)(wcrow + k);
        accB = __builtin_amdgcn_wmma_f32_16x16x4_f32(false, a, false, bb, (short)0, accB, false, false);
        accC = __builtin_amdgcn_wmma_f32_16x16x4_f32(false, a, false, bc, (short)0, accC, false, false);
    }

    const int rbase = m0 + ((lane >> 4) * 8);
#pragma unroll
    for (int r = 0; r < 8; ++r) {
        Bin[(size_t)(rbase + r) * NST + l15] = accB[r];
        Cin[(size_t)(rbase + r) * NST + l15] = accC[r];
    }
}

// ---------------------------------------------------------------------------
// Chunked parallel scan: h_t = a_t*h_{t-1} + u_t, a_t = exp2(dt*a2[n]),
// u_t = dt*B_t[n]*xn. Pass 1: per-chunk product P and zero-state response Q.
// ---------------------------------------------------------------------------
__global__ __launch_bounds__(1024) void scan_pass1(
    const float* __restrict__ dt, const float* __restrict__ xn,
    const float* __restrict__ Bin, const float* __restrict__ A_log,
    float* __restrict__ P, float* __restrict__ Hloc) {
    const int b = blockIdx.x / CCH;
    const int c = blockIdx.x % CCH;
    const int d = threadIdx.x;

    float a2[NST], h[NST], p[NST];
#pragma unroll
    for (int n = 0; n < NST; ++n) {
        a2[n] = -__expf(A_log[(size_t)d * NST + n]) * 1.4426950408889634f;
        h[n] = 0.f;
        p[n] = 1.f;
    }
    const size_t row0 = (size_t)b * LSEQ + (size_t)c * TCH;
    for (int t = 0; t < TCH; ++t) {
        const size_t row = row0 + t;
        const float dtv = dt[row * DDIM + d];
        const float dx  = dtv * xn[row * DDIM + d];
        const float* br = Bin + row * NST;   // block-uniform address -> scalar loads
#pragma unroll
        for (int n = 0; n < NST; ++n) {
            const float ab = __builtin_amdgcn_exp2f(dtv * a2[n]);
            h[n] = fmaf(ab, h[n], dx * br[n]);
            p[n] *= ab;
        }
    }
    float* Pp = P    + ((size_t)blockIdx.x * DDIM + d) * NST;
    float* Hp = Hloc + ((size_t)blockIdx.x * DDIM + d) * NST;
#pragma unroll
    for (int i = 0; i < 4; ++i) {
        ((float4*)Pp)[i] = make_float4(p[4 * i], p[4 * i + 1], p[4 * i + 2], p[4 * i + 3]);
        ((float4*)Hp)[i] = make_float4(h[4 * i], h[4 * i + 1], h[4 * i + 2], h[4 * i + 3]);
    }
}

// Pass 2: serial fix-up over the 32 chunks; rewrites Hloc[c] with the
// corrected INITIAL state of chunk c. One lane per (b, d, n); coalesced.
__global__ __launch_bounds__(1024) void scan_pass2(
    const float* __restrict__ P, float* __restrict__ Hloc) {
    const int tid = blockIdx.x * 1024 + threadIdx.x;  // 0 .. B*D*N-1
    const int b   = tid >> 14;                        // / (DDIM*NST)
    const int idx = tid & 16383;
    float H = 0.f;
    for (int c = 0; c < CCH; ++c) {
        const size_t base = ((size_t)(b * CCH + c) << 14) + idx;
        const float p = P[base];
        const float q = Hloc[base];
        Hloc[base] = H;                // initial state for chunk c
        H = fmaf(p, H, q);
    }
}

// Pass 3: re-run each chunk from its corrected initial state; emit y + residual.
__global__ __launch_bounds__(1024) void scan_pass3(
    const float* __restrict__ x, const float* __restrict__ xn,
    const float* __restrict__ dt, const float* __restrict__ Bin,
    const float* __restrict__ Cin, const float* __restrict__ A_log,
    const float* __restrict__ Dp, const float* __restrict__ Hinit,
    float* __restrict__ y) {
    const int b = blockIdx.x / CCH;
    const int c = blockIdx.x % CCH;
    const int d = threadIdx.x;

    float a2[NST], h[NST];
#pragma unroll
    for (int n = 0; n < NST; ++n)
        a2[n] = -__expf(A_log[(size_t)d * NST + n]) * 1.4426950408889634f;

    const float4* Hp = (const float4*)(Hinit + ((size_t)blockIdx.x * DDIM + d) * NST);
#pragma unroll
    for (int i = 0; i < 4; ++i) {
        const float4 v = Hp[i];
        h[4 * i] = v.x; h[4 * i + 1] = v.y; h[4 * i + 2] = v.z; h[4 * i + 3] = v.w;
    }
    const float dp    = Dp[d];
    const size_t row0 = (size_t)b * LSEQ + (size_t)c * TCH;
    for (int t = 0; t < TCH; ++t) {
        const size_t row = row0 + t;
        const float dtv = dt[row * DDIM + d];
        const float dx  = dtv * xn[row * DDIM + d];
        const float* br = Bin + row * NST;
        const float* cr = Cin + row * NST;
        float yv = 0.f;
#pragma unroll
        for (int n = 0; n < NST; ++n) {
            const float ab = __builtin_amdgcn_exp2f(dtv * a2[n]);
            h[n] = fmaf(ab, h[n], dx * br[n]);
            yv   = fmaf(cr[n], h[n], yv);
        }
        y[row * DDIM + d] = fmaf(dp, x[row * DDIM + d], yv);
    }
}

// ---------------------------------------------------------------------------
// Host launcher
// ---------------------------------------------------------------------------
extern "C" void kernel_launch(void* const* d_in, const int* in_sizes, int n_in,
                              void* d_out, int out_size, void* d_ws, size_t ws_size,
                              hipStream_t stream) {
    const float* x     = (const float*)d_in[0];
    const float* W_dt  = (const float*)d_in[1];
    const float* b_dt  = (const float*)d_in[2];
    const float* W_B   = (const float*)d_in[3];
    const float* W_C   = (const float*)d_in[4];
    const float* D_p   = (const float*)d_in[5];
    const float* A_log = (const float*)d_in[6];
    const float* gamma = (const float*)d_in[7];
    const float* beta  = (const float*)d_in[8];
    float* y = (float*)d_out;

    // Workspace: xn(16MB) | dt(16MB) | Bin(.25MB) | Cin(.25MB) | P(4MB) | Hloc(4MB)
    float* xn   = (float*)d_ws;
    float* dt   = xn + (size_t)MROWS * DDIM;
    float* Bin  = dt + (size_t)MROWS * DDIM;
    float* Cin  = Bin + (size_t)MROWS * NST;
    float* P    = Cin + (size_t)MROWS * NST;
    float* Hloc = P + (size_t)BSZ * CCH * DDIM * NST;

    ln_kernel<<<MROWS, 256, 0, stream>>>(x, gamma, beta, xn);
#if HAVE_TDM
    gemm_dt_tdm_kernel<<<dim3(DDIM / 64, MROWS / 128), 256, 0, stream>>>(xn, W_dt, b_dt, dt);
#else
    gemm_dt_direct_kernel<<<dim3(DDIM / 64, MROWS / 128), 256, 0, stream>>>(xn, W_dt, b_dt, dt);
#endif
    gemm_bc_kernel<<<MROWS / 16 / 8, 256, 0, stream>>>(xn, W_B, W_C, Bin, Cin);

    scan_pass1<<<BSZ * CCH, 1024, 0, stream>>>(dt, xn, Bin, A_log, P, Hloc);
    scan_pass2<<<(BSZ * DDIM * NST) / 1024, 1024, 0, stream>>>(P, Hloc);
    scan_pass3<<<BSZ * CCH, 1024, 0, stream>>>(x, xn, dt, Bin, Cin, A_log, D_p, Hloc, y);
}